// LSTM_33320356282740
// MI455X (gfx1250) — compile-verified
//
#include <hip/hip_runtime.h>

#define B_   128
#define T_   512
#define F_   128
#define H_   1024
#define O_   128
#define PRE_ 24
#define G4H  (4 * H_)

typedef float v2f __attribute__((ext_vector_type(2)));
typedef float v8f __attribute__((ext_vector_type(8)));

#define KC 64   // K-chunk staged in LDS
#define LP 68   // padded LDS row stride in floats (68%64=4 -> conflict-free b64 reads, 16B-aligned rows)

__device__ __forceinline__ float fast_sig(float x) {
    return __builtin_amdgcn_rcpf(1.0f + __expf(-x));
}
__device__ __forceinline__ float fast_tanh(float x) {
    return 2.0f * __builtin_amdgcn_rcpf(1.0f + __expf(-2.0f * x)) - 1.0f;
}

// Register-resident prefetch buffer for one K-chunk (A: 128x64, W: 64x64 floats per block)
struct Frag {
    float4 a[8];
    float4 w[4];
};

__device__ __forceinline__ void chunk_params(int c, int nch1, long lda1, int K1,
                                             const float* A1, const float* W1,
                                             const float* A2, const float* W2,
                                             const float*& Ap, const float*& Wp,
                                             long& lda, int& K, int& k0)
{
    if (c < nch1) { Ap = A1; Wp = W1; lda = lda1;     K = K1; k0 = c * KC; }
    else          { Ap = A2; Wp = W2; lda = (long)H_; K = H_; k0 = (c - nch1) * KC; }
}

__device__ __forceinline__ void load_frag(Frag& f, int tid, int n0,
                                          const float* __restrict__ Ap,
                                          const float* __restrict__ Wp,
                                          long lda, int K, int k0)
{
    #pragma unroll
    for (int i = 0; i < 8; ++i) {                 // A chunk: 128 rows x 16 float4
        int idx = tid + i * 256;
        int row = idx >> 4, j = idx & 15;
        f.a[i] = *(const float4*)(Ap + (long)row * lda + k0 + j * 4);
    }
    #pragma unroll
    for (int i = 0; i < 4; ++i) {                 // W chunk: 64 rows (g*16+n) x 16 float4
        int idx = tid + i * 256;
        int wr  = idx >> 4, j = idx & 15;
        int g   = wr >> 4,  n = wr & 15;
        f.w[i] = *(const float4*)(Wp + (long)(g * H_ + n0 + n) * K + k0 + j * 4);
    }
}

__device__ __forceinline__ void store_frag(const Frag& f, int tid,
                                           float* __restrict__ lds_a,
                                           float* __restrict__ lds_w)
{
    #pragma unroll
    for (int i = 0; i < 8; ++i) {
        int idx = tid + i * 256;
        int row = idx >> 4, j = idx & 15;
        *(float4*)(lds_a + row * LP + j * 4) = f.a[i];
    }
    #pragma unroll
    for (int i = 0; i < 4; ++i) {
        int idx = tid + i * 256;
        int wr  = idx >> 4, j = idx & 15;
        *(float4*)(lds_w + wr * LP + j * 4) = f.w[i];
    }
}

// Fused LSTM cell step for one layer:
//   gates = A1 @ W1^T + A2 @ W2^T + beff   (gate g lives in rows [g*H, (g+1)*H) of W)
//   c' = sig(f)*c + sig(i)*tanh(g);  h' = sig(o)*tanh(c')
// Grid: H_/16 blocks (n-tiles), 256 threads = 8 waves (m-tiles over B=128).
__global__ __launch_bounds__(256)
void lstm_step(const float* __restrict__ A1, long lda1, int K1,
               const float* __restrict__ W1,     // [4H x K1] row-major
               const float* __restrict__ A2,     // [B x H], lda = H
               const float* __restrict__ W2,     // [4H x H] row-major
               const float* __restrict__ beff,   // [4H]
               float* __restrict__ Cst,          // [B x H] in/out cell state
               float* __restrict__ Hout,         // [B x H] new hidden
               float* __restrict__ Hsto)         // [B x H] optional copy (may be null)
{
    __shared__ float lds_a[B_ * LP];   // 128 rows x KC (padded)
    __shared__ float lds_w[64 * LP];   // 4 gates x 16 n x KC (padded)

    const int tid  = threadIdx.x;
    const int wave = tid >> 5;
    const int lane = tid & 31;
    const int hi   = lane >> 4;   // lane half: selects K pair (A/B) and M+8 (C/D)
    const int ln   = lane & 15;
    const int n0   = blockIdx.x * 16;
    const int m0   = wave * 16;

    v8f acc0 = (v8f)0.0f, acc1 = (v8f)0.0f, acc2 = (v8f)0.0f, acc3 = (v8f)0.0f;

    const int nch1 = K1 / KC;
    const int nch  = nch1 + H_ / KC;

    const float *Ap, *Wp; long lda; int K, k0;
    Frag f;
    chunk_params(0, nch1, lda1, K1, A1, W1, A2, W2, Ap, Wp, lda, K, k0);
    load_frag(f, tid, n0, Ap, Wp, lda, K, k0);

    for (int c = 0; c < nch; ++c) {
        __syncthreads();                           // previous chunk's compute done
        store_frag(f, tid, lds_a, lds_w);          // implicit wait on in-flight global loads
        __syncthreads();

        // prefetch next chunk into registers; overlaps the WMMA burst below
        if (c + 1 < nch) {
            chunk_params(c + 1, nch1, lda1, K1, A1, W1, A2, W2, Ap, Wp, lda, K, k0);
            load_frag(f, tid, n0, Ap, Wp, lda, K, k0);
        }

        // ---- WMMA f32 16x16x4 over this K-chunk; A fragment reused across 4 gates ----
        const float* ap = lds_a + (m0 + ln) * LP + 2 * hi;
        const float* wp = lds_w + ln * LP + 2 * hi;
        #pragma unroll
        for (int kk = 0; kk < KC; kk += 4) {
            v2f a  = *(const v2f*)(ap + kk);
            v2f b0 = *(const v2f*)(wp + kk);
            v2f b1 = *(const v2f*)(wp + 16 * LP + kk);
            v2f b2 = *(const v2f*)(wp + 32 * LP + kk);
            v2f b3 = *(const v2f*)(wp + 48 * LP + kk);
            acc0 = __builtin_amdgcn_wmma_f32_16x16x4_f32(false, a, false, b0, (short)0, acc0, false, false);
            acc1 = __builtin_amdgcn_wmma_f32_16x16x4_f32(false, a, false, b1, (short)0, acc1, false, false);
            acc2 = __builtin_amdgcn_wmma_f32_16x16x4_f32(false, a, false, b2, (short)0, acc2, false, false);
            acc3 = __builtin_amdgcn_wmma_f32_16x16x4_f32(false, a, false, b3, (short)0, acc3, false, false);
        }
    }

    // ---- epilogue: bias, activations, cell/hidden update ----
    const int col = n0 + ln;
    const float bi = beff[0 * H_ + col];
    const float bf = beff[1 * H_ + col];
    const float bg = beff[2 * H_ + col];
    const float bo = beff[3 * H_ + col];
    #pragma unroll
    for (int r = 0; r < 8; ++r) {
        const int  row = m0 + r + 8 * hi;           // C/D layout: VGPR r -> M=r (+8 for hi lanes)
        const long off = (long)row * H_ + col;
        const float ig = fast_sig(acc0[r] + bi);
        const float fg = fast_sig(acc1[r] + bf);
        const float gg = fast_tanh(acc2[r] + bg);
        const float og = fast_sig(acc3[r] + bo);
        const float cn = fg * Cst[off] + ig * gg;
        const float hn = og * fast_tanh(cn);
        Cst[off]  = cn;
        Hout[off] = hn;
        if (Hsto) Hsto[off] = hn;
    }
}

// W_eff0[r][c] = sum_k W_ih0[r][k] * W_hid[k][c]   ([4096x1024]@[1024x128])
__global__ __launch_bounds__(256)
void weff_kernel(const float* __restrict__ Wih0, const float* __restrict__ Whid,
                 float* __restrict__ Weff)
{
    __shared__ float As[16][17];
    __shared__ float Bs[16][17];
    const int tx = threadIdx.x, ty = threadIdx.y;
    const int r0 = blockIdx.y * 16;
    const int c0 = blockIdx.x * 16;
    float acc = 0.f;
    for (int kb = 0; kb < H_; kb += 16) {
        As[ty][tx] = Wih0[(long)(r0 + ty) * H_ + kb + tx];
        Bs[ty][tx] = Whid[(long)(kb + ty) * F_ + c0 + tx];
        __syncthreads();
        #pragma unroll
        for (int k = 0; k < 16; ++k) acc += As[ty][k] * Bs[k][tx];
        __syncthreads();
    }
    Weff[(long)(r0 + ty) * F_ + c0 + tx] = acc;
}

// beff0 = b_ih[0] + b_hh[0] + W_ih0 @ b_hid ;  beff1 = b_ih[1] + b_hh[1]
__global__ void beff_kernel(const float* __restrict__ Wih0, const float* __restrict__ bhid,
                            const float* __restrict__ bih, const float* __restrict__ bhh,
                            float* __restrict__ beff0, float* __restrict__ beff1)
{
    int r = blockIdx.x * blockDim.x + threadIdx.x;
    if (r >= G4H) return;
    float d = 0.f;
    for (int k = 0; k < H_; ++k) d += Wih0[(long)r * H_ + k] * bhid[k];
    beff0[r] = bih[r] + bhh[r] + d;
    beff1[r] = bih[G4H + r] + bhh[G4H + r];
}

// out[b][tt][o] = relu(Hsto[tt][b][:]) . W_out[o][:] + b_out[o]
__global__ __launch_bounds__(256)
void out_kernel(const float* __restrict__ Hsto, const float* __restrict__ Wout,
                const float* __restrict__ bout, float* __restrict__ out)
{
    __shared__ float As[16][17];
    __shared__ float Bs[16][17];
    const int tx = threadIdx.x, ty = threadIdx.y;
    const int m0 = blockIdx.y * 16;              // m = b*PRE_ + tt, M = 3072
    const int c0 = blockIdx.x * 16;
    const int m  = m0 + ty;
    const int b  = m / PRE_;
    const int tt = m % PRE_;
    const long arow = ((long)tt * B_ + b) * H_;
    float acc = 0.f;
    for (int kb = 0; kb < H_; kb += 16) {
        float av = Hsto[arow + kb + tx];
        As[ty][tx] = av > 0.f ? av : 0.f;
        Bs[tx][ty] = Wout[(long)(c0 + ty) * H_ + kb + tx];   // Bs[k][o]
        __syncthreads();
        #pragma unroll
        for (int k = 0; k < 16; ++k) acc += As[ty][k] * Bs[k][tx];
        __syncthreads();
    }
    out[(long)m * O_ + c0 + tx] = acc + bout[c0 + tx];
}

__global__ void zero_kernel(float* __restrict__ p, long n)
{
    long i = (long)blockIdx.x * blockDim.x + threadIdx.x;
    if (i < n) p[i] = 0.f;
}

extern "C" void kernel_launch(void* const* d_in, const int* in_sizes, int n_in,
                              void* d_out, int out_size, void* d_ws, size_t ws_size,
                              hipStream_t stream)
{
    const float* x     = (const float*)d_in[0];
    const float* W_hid = (const float*)d_in[1];
    const float* b_hid = (const float*)d_in[2];
    const float* W_ih  = (const float*)d_in[3];   // [2, 4096, 1024]
    const float* W_hh  = (const float*)d_in[4];   // [2, 4096, 1024]
    const float* b_ih  = (const float*)d_in[5];   // [2, 4096]
    const float* b_hh  = (const float*)d_in[6];
    const float* W_out = (const float*)d_in[7];
    const float* b_out = (const float*)d_in[8];
    float* out = (float*)d_out;

    const long BH = (long)B_ * H_;
    float* ws    = (float*)d_ws;
    float* weff0 = ws;                         // 4096*128
    float* beff0 = weff0 + (long)G4H * F_;     // 4096
    float* beff1 = beff0 + G4H;                // 4096
    float* h0    = beff1 + G4H;                // 2*BH (double buffer)
    float* h1    = h0 + 2 * BH;                // 2*BH
    float* c0    = h1 + 2 * BH;                // BH
    float* c1    = c0 + BH;                    // BH
    float* hsto  = c1 + BH;                    // PRE_*BH

    // zero-init h0/h1 (both buffers) + c0/c1 in one shot (contiguous 6*BH)
    const long nz = 6 * BH;
    zero_kernel<<<(int)((nz + 255) / 256), 256, 0, stream>>>(h0, nz);

    // precompute fused layer-0 weights/bias
    weff_kernel<<<dim3(F_ / 16, G4H / 16), dim3(16, 16), 0, stream>>>(W_ih, W_hid, weff0);
    beff_kernel<<<G4H / 256, 256, 0, stream>>>(W_ih, b_hid, b_ih, b_hh, beff0, beff1);

    const float* Wih1 = W_ih + (long)G4H * H_;
    const float* Whh0 = W_hh;
    const float* Whh1 = W_hh + (long)G4H * H_;

    // Only steps 0..T-2 are observable (h_seq[:-1]); store h1 for t in [T-1-PRE, T-2].
    for (int t = 0; t < T_ - 1; ++t) {
        const int pb = t & 1, cb = 1 - pb;
        // layer 0: gates = x_t @ Weff0^T + h0 @ Whh0^T + beff0
        lstm_step<<<H_ / 16, 256, 0, stream>>>(
            x + (long)t * F_, (long)T_ * F_, F_,
            weff0,
            h0 + pb * BH, Whh0, beff0,
            c0, h0 + cb * BH, nullptr);
        // layer 1: gates = h0' @ Wih1^T + h1 @ Whh1^T + beff1
        float* hs = (t >= T_ - 1 - PRE_) ? (hsto + (long)(t - (T_ - 1 - PRE_)) * BH) : nullptr;
        lstm_step<<<H_ / 16, 256, 0, stream>>>(
            h0 + cb * BH, (long)H_, H_,
            Wih1,
            h1 + pb * BH, Whh1, beff1,
            c1, h1 + cb * BH, hs);
    }

    out_kernel<<<dim3(O_ / 16, (B_ * PRE_) / 16), dim3(16, 16), 0, stream>>>(
        hsto, W_out, b_out, out);
}